// YOLOv3_7696581394896
// MI455X (gfx1250) — compile-verified
//
#include <hip/hip_runtime.h>
#include <math.h>

// ---------------- constants from the reference ----------------
#define NB   32
#define NA   3
#define NCLS 80
#define NCH  (5 + NCLS)      // 85
#define NG   64
#define STRIDE_ 32
#define MAXGT 50
#define IGNORE_THRE 0.6f

typedef __attribute__((ext_vector_type(2))) float v2f;
typedef __attribute__((ext_vector_type(8))) float v8f;

__device__ __forceinline__ float softplusf(float x) {
    // jax.nn.softplus(x) = max(x,0) + log1p(exp(-|x|))
    return fmaxf(x, 0.0f) + log1pf(expf(-fabsf(x)));
}
__device__ __forceinline__ float sigmoidf_(float x) {
    return 1.0f / (1.0f + expf(-x));
}
__device__ __forceinline__ float iou_xywh(float px, float py, float pw, float ph,
                                          float gx, float gy, float gw, float gh) {
    float tlx = fmaxf(px - pw * 0.5f, gx - gw * 0.5f);
    float tly = fmaxf(py - ph * 0.5f, gy - gh * 0.5f);
    float brx = fminf(px + pw * 0.5f, gx + gw * 0.5f);
    float bry = fminf(py + ph * 0.5f, gy + gh * 0.5f);
    float w = fmaxf(brx - tlx, 0.0f);
    float h = fmaxf(bry - tly, 0.0f);
    float inter = w * h;
    return inter / (pw * ph + gw * gh - inter + 1e-16f);
}

// ---------------- kernel 1: per-GT prep (tiny) ----------------
// ws entry per (b,t): 12 dwords
// [0..3] gx,gy,gw,gh  [4..8] tx,ty,tw,th,wv  [9] cell(int,-1=invalid) [10] cls(int) [11] gt_valid(int)
__global__ void yolo_prep(const float* __restrict__ labels,
                          const int*   __restrict__ num_gt,
                          const int*   __restrict__ img_size_p,
                          float*       __restrict__ ws) {
    const int b = blockIdx.x;
    const int t = threadIdx.x;
    if (t >= MAXGT) return;

    const float AW[9] = {10.f, 16.f, 33.f, 30.f, 62.f, 59.f, 116.f, 156.f, 373.f};
    const float AH[9] = {13.f, 30.f, 23.f, 61.f, 45.f, 119.f, 90.f, 198.f, 326.f};

    const float img = (float)img_size_p[0];
    const float* L = labels + ((size_t)b * MAXGT + t) * 5;
    const float cls = L[0];
    const float gx = L[1], gy = L[2], gw = L[3], gh = L[4];
    const int gt_valid = (t < num_gt[b]) ? 1 : 0;

    // anchor iou argmax (first-max wins like jnp.argmax)
    const float area_g = gw * gh;
    float best = -1.0f; int bestn = 0;
    #pragma unroll
    for (int k = 0; k < 9; ++k) {
        float inter = fminf(gw, AW[k]) * fminf(gh, AH[k]);
        float iou = inter / (area_g + AW[k] * AH[k] - inter + 1e-16f);
        if (iou > best) { best = iou; bestn = k; }
    }
    const bool matched = (bestn >= 6);           // ANCHOR_IDX = {6,7,8}
    const bool valid = matched && (gt_valid != 0);
    const int tn = bestn % NA;
    const float gxs = gx * (1.0f / STRIDE_);
    const float gys = gy * (1.0f / STRIDE_);
    const int ti = (int)gxs;
    const int tj = (int)gys;
    const float tx = gxs - (float)ti;
    const float ty = gys - (float)tj;
    const float aw = AW[6 + tn], ah = AH[6 + tn];
    const float tw = logf(gw / aw + 1e-8f);
    const float th = logf(gh / ah + 1e-8f);
    const float wv = 2.0f - area_g / (img * img);
    const int cell = valid ? (tn * (NG * NG) + tj * NG + ti) : -1;

    float* e = ws + ((size_t)b * MAXGT + t) * 12;
    e[0] = gx; e[1] = gy; e[2] = gw; e[3] = gh;
    e[4] = tx; e[5] = ty; e[6] = tw; e[7] = th; e[8] = wv;
    ((int*)e)[9]  = cell;
    ((int*)e)[10] = (int)cls;
    ((int*)e)[11] = gt_valid;
}

// ---------------- wave/block reduction via f32 WMMA ----------------
// D = ones(16x4) x B(4x16) + 0 : each lane partial lands in exactly one B slot
// (b.vgpr1 = 0), so the 16 column sums contain every lane's value once.
__device__ __forceinline__ float block_reduce_sum(float v, float* s_wsum) {
    v2f a; a[0] = 1.0f; a[1] = 1.0f;
    v2f bm; bm[0] = v; bm[1] = 0.0f;
    v8f c = {};
    c = __builtin_amdgcn_wmma_f32_16x16x4_f32(false, a, false, bm,
                                              (short)0, c, false, false);
    float col = c[0];                      // per-lane column sum (dup in hi half)
    col += __shfl_xor(col, 8);
    col += __shfl_xor(col, 4);
    col += __shfl_xor(col, 2);
    col += __shfl_xor(col, 1);             // lane 0 (and 16): wave total
    const int lane = threadIdx.x & 31;
    const int wave = threadIdx.x >> 5;
    if (lane == 0) s_wsum[wave] = col;
    __syncthreads();
    float total = 0.0f;
    if (threadIdx.x == 0) {
        #pragma unroll
        for (int w = 0; w < 8; ++w) total += s_wsum[w];
    }
    return total;
}

// ---------------- kernel 2: per-cell losses ----------------
// One block = one (b, a, 4 rows of j). 256 threads, thread = one cell (j,i),
// i contiguous -> coalesced per-channel-plane loads.
__global__ void yolo_loss(const float* __restrict__ raw,
                          const float* __restrict__ ws,
                          float* __restrict__ out) {
    __shared__ float s_box[MAXGT][4];
    __shared__ float s_tv[MAXGT][5];   // tx,ty,tw,th,wv
    __shared__ int   s_cell[MAXGT];
    __shared__ int   s_cls[MAXGT];
    __shared__ int   s_gtv[MAXGT];
    __shared__ int   s_hasvalid;
    __shared__ float s_wsum[8];

    const float ANW[NA] = {116.f, 156.f, 373.f};
    const float ANH[NA] = {90.f, 198.f, 326.f};

    const int bi  = blockIdx.x;
    const int b   = bi / (NA * (NG / 4));
    const int rem = bi % (NA * (NG / 4));
    const int a   = rem / (NG / 4);
    const int j0  = (rem % (NG / 4)) * 4;
    const int tid = threadIdx.x;

    if (tid < MAXGT) {
        const float* e = ws + ((size_t)b * MAXGT + tid) * 12;
        s_box[tid][0] = e[0]; s_box[tid][1] = e[1];
        s_box[tid][2] = e[2]; s_box[tid][3] = e[3];
        s_tv[tid][0] = e[4]; s_tv[tid][1] = e[5]; s_tv[tid][2] = e[6];
        s_tv[tid][3] = e[7]; s_tv[tid][4] = e[8];
        s_cell[tid] = ((const int*)e)[9];
        s_cls[tid]  = ((const int*)e)[10];
        s_gtv[tid]  = ((const int*)e)[11];
    }
    __syncthreads();
    if (tid == 0) {
        int hv = 0;
        for (int t = 0; t < MAXGT; ++t) hv |= (s_cell[t] >= 0);
        s_hasvalid = hv;
    }
    __syncthreads();

    const int i  = tid & (NG - 1);
    const int jj = j0 + (tid >> 6);

    // channel plane stride = NG*NG floats
    const float* cellp = raw +
        ((size_t)(b * (NA * NCH) + a * NCH)) * (NG * NG) + (size_t)jj * NG + i;

    const float p0 = cellp[0 * NG * NG];
    const float p1 = cellp[1 * NG * NG];
    const float p2 = cellp[2 * NG * NG];
    const float p3 = cellp[3 * NG * NG];
    const float p4 = cellp[4 * NG * NG];

    // predicted box
    const float px = (sigmoidf_(p0) + (float)i)  * (float)STRIDE_;
    const float py = (sigmoidf_(p1) + (float)jj) * (float)STRIDE_;
    const float pw = expf(p2) * ANW[a];
    const float ph = expf(p3) * ANH[a];

    // max IoU against valid GTs (gt_valid mask, not the stricter 'valid')
    float maxiou = -1.0f;
    for (int t = 0; t < MAXGT; ++t) {
        if (s_gtv[t]) {
            float u = iou_xywh(px, py, pw, ph,
                               s_box[t][0], s_box[t][1], s_box[t][2], s_box[t][3]);
            maxiou = fmaxf(maxiou, u);
        }
    }

    // target lookup: last valid GT scattered to this cell wins
    const int mycell = a * (NG * NG) + jj * NG + i;
    int tgt = -1;
    for (int t = 0; t < MAXGT; ++t)
        if (s_cell[t] == mycell) tgt = t;
    const bool m = (tgt >= 0);

    bool keep = s_hasvalid ? (maxiou < IGNORE_THRE) : true;
    keep = keep || m;

    float acc = 0.0f;
    if (keep) acc += m ? (softplusf(p4) - p4) : softplusf(p4);   // bce(conf, m?1:0)

    if (m) {
        const float tx = s_tv[tgt][0], ty = s_tv[tgt][1];
        const float tw = s_tv[tgt][2], th = s_tv[tgt][3];
        const float wv = s_tv[tgt][4];
        acc += wv * ((softplusf(p0) - p0 * tx) + (softplusf(p1) - p1 * ty));
        const float dw = p2 - tw, dh = p3 - th;
        acc += 0.5f * wv * (dw * dw + dh * dh);
        const int cls = s_cls[tgt];
        float csum = 0.0f;
        for (int c = 0; c < NCLS; ++c) {
            const float pc = cellp[(size_t)(5 + c) * NG * NG];
            csum += softplusf(pc);
            if (c == cls) csum -= pc;     // bce(x, onehot) = softplus(x) - x*t
        }
        acc += csum;
    }

    // uniform path: WMMA-based reduction (EXEC all ones)
    const float total = block_reduce_sum(acc, s_wsum);
    if (tid == 0) atomicAdd(out, total);
}

extern "C" void kernel_launch(void* const* d_in, const int* in_sizes, int n_in,
                              void* d_out, int out_size, void* d_ws, size_t ws_size,
                              hipStream_t stream) {
    const float* raw      = (const float*)d_in[0];
    const float* labels   = (const float*)d_in[1];
    const int*   num_gt   = (const int*)d_in[2];
    const int*   img_size = (const int*)d_in[3];
    float*       out      = (float*)d_out;
    float*       ws       = (float*)d_ws;   // uses 32*50*12*4 = 76.8 KB

    hipMemsetAsync(out, 0, sizeof(float) * (size_t)out_size, stream);
    yolo_prep<<<NB, 64, 0, stream>>>(labels, num_gt, img_size, ws);
    yolo_loss<<<NB * NA * (NG / 4), 256, 0, stream>>>(raw, ws, out);
}